// GNNMessagePassing_9131100471786
// MI455X (gfx1250) — compile-verified
//
#include <hip/hip_runtime.h>

typedef __attribute__((ext_vector_type(2))) float v2f;
typedef __attribute__((ext_vector_type(8))) float v8f;

#define TPB 256

// ---------------------------------------------------------------------------
// 1. Stage triplet features: t' = t + ec[corr]/cnt[corr]  (into d_out regions)
// ---------------------------------------------------------------------------
__global__ __launch_bounds__(TPB) void k_stage_tri(
    const float* __restrict__ ec, const float* __restrict__ t12,
    const float* __restrict__ t13, const float* __restrict__ t23,
    const int* __restrict__ c12, const int* __restrict__ c13,
    const int* __restrict__ c23, const float* __restrict__ cnt,
    float* __restrict__ s12, float* __restrict__ s13, float* __restrict__ s23,
    int T) {
  int i = blockIdx.x * blockDim.x + threadIdx.x;
  if (i >= T) return;
  int a = c12[i], b = c13[i], c = c23[i];
  s12[i] = t12[i] + ec[a] / cnt[a];
  s13[i] = t13[i] + ec[b] / cnt[b];
  s23[i] = t23[i] + ec[c] / cnt[c];
}

__global__ __launch_bounds__(TPB) void k_fill(float* __restrict__ p, float v, int n) {
  int i = blockIdx.x * blockDim.x + threadIdx.x;
  if (i < n) p[i] = v;
}

// deg[dst] += 1  (indegree; deg pre-filled with 1.0 for the self-loop)
__global__ __launch_bounds__(TPB) void k_deg_count(const int* __restrict__ dst,
                                                   float* __restrict__ deg, int M) {
  int e = blockIdx.x * blockDim.x + threadIdx.x;
  if (e < M) atomicAdd(&deg[dst[e]], 1.0f);
}

__global__ __launch_bounds__(TPB) void k_rsqrt_inplace(float* __restrict__ deg, int T) {
  int i = blockIdx.x * blockDim.x + threadIdx.x;
  if (i < T) deg[i] = rsqrtf(deg[i]);
}

// ---------------------------------------------------------------------------
// Layer-1 aggregation in 3-dim space (N(A)X, then @W1 afterwards)
// ---------------------------------------------------------------------------
__global__ __launch_bounds__(TPB) void k_selfinit3(
    const float* __restrict__ s12, const float* __restrict__ s13,
    const float* __restrict__ s23, const float* __restrict__ dis,
    float* __restrict__ agg3, int T) {
  int i = blockIdx.x * blockDim.x + threadIdx.x;
  if (i >= T) return;
  float w = dis[i] * dis[i];
  agg3[3 * i + 0] = s12[i] * w;
  agg3[3 * i + 1] = s13[i] * w;
  agg3[3 * i + 2] = s23[i] * w;
}

__global__ __launch_bounds__(TPB) void k_scatter3(
    const int* __restrict__ ei, const float* __restrict__ s12,
    const float* __restrict__ s13, const float* __restrict__ s23,
    const float* __restrict__ dis, float* __restrict__ agg3, int M) {
  int e = blockIdx.x * blockDim.x + threadIdx.x;
  if (e >= M) return;
  int s = ei[e], d = ei[M + e];
  float nrm = dis[s] * dis[d];
  atomicAdd(&agg3[3 * d + 0], s12[s] * nrm);
  atomicAdd(&agg3[3 * d + 1], s13[s] * nrm);
  atomicAdd(&agg3[3 * d + 2], s23[s] * nrm);
}

// h1 = relu(agg3 @ W1 + b1)   (K=3: trivial VALU GEMM; 8 threads / node)
__global__ __launch_bounds__(TPB) void k_h1(
    const float* __restrict__ agg3, const float* __restrict__ W1,
    const float* __restrict__ b1, float* __restrict__ h1, int T) {
  int tid = blockIdx.x * blockDim.x + threadIdx.x;
  if (tid >= T * 8) return;
  int i = tid >> 3;
  int f = (tid & 7) << 2;
  float a0 = agg3[3 * i + 0], a1 = agg3[3 * i + 1], a2 = agg3[3 * i + 2];
  float4 o;
  float* op = &o.x;
#pragma unroll
  for (int k = 0; k < 4; ++k) {
    int ff = f + k;
    float v = b1[ff] + a0 * W1[ff] + a1 * W1[32 + ff] + a2 * W1[64 + ff];
    op[k] = v > 0.f ? v : 0.f;
  }
  *(float4*)(h1 + (size_t)i * 32 + f) = o;
}

// ---------------------------------------------------------------------------
// Layer-2 aggregation: aggH = N(A) h1   (float4 chunks, fp32 global atomics)
// ---------------------------------------------------------------------------
__global__ __launch_bounds__(TPB) void k_selfinit32(
    const float* __restrict__ h1, const float* __restrict__ dis,
    float* __restrict__ aggH, int T) {
  int tid = blockIdx.x * blockDim.x + threadIdx.x;
  if (tid >= T * 8) return;
  int i = tid >> 3;
  int f = (tid & 7) << 2;
  float w = dis[i] * dis[i];
  float4 v = *(const float4*)(h1 + (size_t)i * 32 + f);
  v.x *= w; v.y *= w; v.z *= w; v.w *= w;
  *(float4*)(aggH + (size_t)i * 32 + f) = v;
}

__global__ __launch_bounds__(TPB) void k_scatter32(
    const int* __restrict__ ei, const float* __restrict__ h1,
    const float* __restrict__ dis, float* __restrict__ aggH, int M) {
  int tid = blockIdx.x * blockDim.x + threadIdx.x;
  if (tid >= M * 8) return;
  int e = tid >> 3;
  int f = (tid & 7) << 2;
  int s = ei[e], d = ei[M + e];
  float nrm = dis[s] * dis[d];
  float4 v = *(const float4*)(h1 + (size_t)s * 32 + f);
  float* dp = aggH + (size_t)d * 32 + f;
  atomicAdd(dp + 0, v.x * nrm);
  atomicAdd(dp + 1, v.y * nrm);
  atomicAdd(dp + 2, v.z * nrm);
  atomicAdd(dp + 3, v.w * nrm);
}

// ---------------------------------------------------------------------------
// h2 = relu(aggH @ W2 + b2)  via V_WMMA_F32_16X16X4_F32 (fp32-exact WMMA)
// One wave owns 16 rows; two 16x16 N-tiles; K=32 -> 8 k-steps -> 16 wmma/wave.
// A 16x4 f32 layout: lane<16 holds M=lane, K={4j,4j+1}; lane>=16 K={4j+2,4j+3}
// B 4x16  f32 layout: lane<16 holds N=lane, K={4j,4j+1}; lane>=16 K={4j+2,4j+3}
// C 16x16 f32 layout: vgpr r -> M = r + 8*(lane>=16), N = lane&15
// ---------------------------------------------------------------------------
__global__ __launch_bounds__(TPB) void k_h2_wmma(
    const float* __restrict__ aggH, const float* __restrict__ W2,
    const float* __restrict__ b2, float* __restrict__ h2, int T) {
  const int lane = threadIdx.x & 31;
  const int wave = blockIdx.x * (blockDim.x >> 5) + (threadIdx.x >> 5);
  int r0 = wave * 16;
  if (r0 + 16 > T) r0 = (T >= 16) ? (T - 16) : 0;  // clamp: keep EXEC all-ones
  const int l15 = lane & 15;
  const int hi = lane >> 4;

  v8f c0 = {};
  v8f c1 = {};
#pragma unroll
  for (int j = 0; j < 8; ++j) {
    const int k0 = 4 * j + 2 * hi;
    v2f a = *(const v2f*)(aggH + (size_t)(r0 + l15) * 32 + k0);
    v2f b0, b1v;
    b0.x  = W2[(size_t)k0 * 32 + l15];
    b0.y  = W2[(size_t)(k0 + 1) * 32 + l15];
    b1v.x = W2[(size_t)k0 * 32 + 16 + l15];
    b1v.y = W2[(size_t)(k0 + 1) * 32 + 16 + l15];
    c0 = __builtin_amdgcn_wmma_f32_16x16x4_f32(false, a, false, b0,
                                               (short)0, c0, false, false);
    c1 = __builtin_amdgcn_wmma_f32_16x16x4_f32(false, a, false, b1v,
                                               (short)0, c1, false, false);
  }
  const float bias0 = b2[l15];
  const float bias1 = b2[16 + l15];
#pragma unroll
  for (int r = 0; r < 8; ++r) {
    const int Mrow = r + 8 * hi;
    float v0 = c0[r] + bias0; v0 = v0 > 0.f ? v0 : 0.f;
    float v1 = c1[r] + bias1; v1 = v1 > 0.f ? v1 : 0.f;
    h2[(size_t)(r0 + Mrow) * 32 + l15]      = v0;
    h2[(size_t)(r0 + Mrow) * 32 + 16 + l15] = v1;
  }
}

// ---------------------------------------------------------------------------
// Head + triplet finalize: delta = h2 @ Wout + bout ; t_out = t' - delta ;
// edge_acc[corr] += delta
// ---------------------------------------------------------------------------
__global__ __launch_bounds__(TPB) void k_final_tri(
    const float* __restrict__ h2, const float* __restrict__ Wout,
    const float* __restrict__ bout, const int* __restrict__ c12,
    const int* __restrict__ c13, const int* __restrict__ c23,
    float* __restrict__ s12, float* __restrict__ s13, float* __restrict__ s23,
    float* __restrict__ acc, int T) {
  int i = blockIdx.x * blockDim.x + threadIdx.x;
  if (i >= T) return;
  float row[32];
  const float4* hp = (const float4*)(h2 + (size_t)i * 32);
#pragma unroll
  for (int j = 0; j < 8; ++j) {
    float4 v = hp[j];
    row[4 * j + 0] = v.x; row[4 * j + 1] = v.y;
    row[4 * j + 2] = v.z; row[4 * j + 3] = v.w;
  }
  float d0 = bout[0], d1 = bout[1], d2 = bout[2];
#pragma unroll
  for (int f = 0; f < 32; ++f) {
    float h = row[f];
    d0 += h * Wout[3 * f + 0];
    d1 += h * Wout[3 * f + 1];
    d2 += h * Wout[3 * f + 2];
  }
  s12[i] = s12[i] - d0;
  s13[i] = s13[i] - d1;
  s23[i] = s23[i] - d2;
  atomicAdd(&acc[c12[i]], d0);
  atomicAdd(&acc[c13[i]], d1);
  atomicAdd(&acc[c23[i]], d2);
}

__global__ __launch_bounds__(TPB) void k_final_edges(
    const float* __restrict__ ec, const float* __restrict__ cnt,
    const float* __restrict__ acc, float* __restrict__ out, int E) {
  int e = blockIdx.x * blockDim.x + threadIdx.x;
  if (e >= E) return;
  float base = (cnt[e] > 0.f) ? 0.f : ec[e];
  out[e] = base + acc[e];
}

// ---------------------------------------------------------------------------
extern "C" void kernel_launch(void* const* d_in, const int* in_sizes, int n_in,
                              void* d_out, int out_size, void* d_ws, size_t ws_size,
                              hipStream_t stream) {
  const float* ec   = (const float*)d_in[0];
  const float* t12  = (const float*)d_in[1];
  const float* t13  = (const float*)d_in[2];
  const float* t23  = (const float*)d_in[3];
  const int*   c12  = (const int*)d_in[4];
  const int*   c13  = (const int*)d_in[5];
  const int*   c23  = (const int*)d_in[6];
  const float* cnt  = (const float*)d_in[7];
  const int*   ei   = (const int*)d_in[8];
  const float* W1   = (const float*)d_in[9];
  const float* b1   = (const float*)d_in[10];
  const float* W2   = (const float*)d_in[11];
  const float* b2   = (const float*)d_in[12];
  const float* Wout = (const float*)d_in[13];
  const float* bout = (const float*)d_in[14];

  const int E = in_sizes[0];
  const int T = in_sizes[1];
  const int M = in_sizes[8] / 2;

  float* out = (float*)d_out;       // [ec(E) | t12(T) | t13(T) | t23(T)]
  float* s12 = out + E;
  float* s13 = out + E + (size_t)T;
  float* s23 = out + E + 2 * (size_t)T;

  // Workspace: bufA(T*32) | bufB(T*32) | dis(T). agg3/edge_acc alias bufB
  // (disjoint lifetimes). Needs (65*T)*4 bytes.
  float* ws   = (float*)d_ws;
  const size_t T32 = (size_t)T * 32;
  float* bufA = ws;              // h1, then h2
  float* bufB = ws + T32;        // agg3 -> aggH -> edge_acc
  float* dis  = ws + 2 * T32;    // deg, then deg^-1/2
  float* agg3 = bufB;
  float* acc  = bufB;

  const int g_T  = (T + TPB - 1) / TPB;
  const int g_E  = (E + TPB - 1) / TPB;
  const int g_M  = (M + TPB - 1) / TPB;
  const int g_T8 = (T * 8 + TPB - 1) / TPB;
  const int g_M8 = (M * 8 + TPB - 1) / TPB;
  const int nwaves = (T + 15) / 16;
  const int g_wmma = (nwaves + (TPB / 32) - 1) / (TPB / 32);

  // 1. stage t' into d_out
  k_stage_tri<<<g_T, TPB, 0, stream>>>(ec, t12, t13, t23, c12, c13, c23, cnt,
                                       s12, s13, s23, T);
  // 2. degree & normalization
  k_fill<<<g_T, TPB, 0, stream>>>(dis, 1.0f, T);
  k_deg_count<<<g_M, TPB, 0, stream>>>(ei + M, dis, M);
  k_rsqrt_inplace<<<g_T, TPB, 0, stream>>>(dis, T);
  // 3. layer 1: aggregate in 3-dim, then @W1 (+b1, relu) -> bufA
  k_selfinit3<<<g_T, TPB, 0, stream>>>(s12, s13, s23, dis, agg3, T);
  k_scatter3<<<g_M, TPB, 0, stream>>>(ei, s12, s13, s23, dis, agg3, M);
  k_h1<<<g_T8, TPB, 0, stream>>>(agg3, W1, b1, bufA, T);
  // 4. layer 2: aggregate h1 -> bufB, then WMMA GEMM (+b2, relu) -> bufA
  k_selfinit32<<<g_T8, TPB, 0, stream>>>(bufA, dis, bufB, T);
  k_scatter32<<<g_M8, TPB, 0, stream>>>(ei, bufA, dis, bufB, M);
  k_h2_wmma<<<g_wmma, TPB, 0, stream>>>(bufB, W2, b2, bufA, T);
  // 5. head + scatter deltas to edges
  k_fill<<<g_E, TPB, 0, stream>>>(acc, 0.0f, E);
  k_final_tri<<<g_T, TPB, 0, stream>>>(bufA, Wout, bout, c12, c13, c23,
                                       s12, s13, s23, acc, T);
  // 6. final edge costs
  k_final_edges<<<g_E, TPB, 0, stream>>>(ec, cnt, acc, out, E);
}